// MaskedEdgeAttention_962072674434
// MI455X (gfx1250) — compile-verified
//
#include <hip/hip_runtime.h>
#include <math.h>

// ---------------------------------------------------------------------------
// MaskedEdgeAttention for MI455X (gfx1250, wave32)
//   S=512 seq, B=64 batch, D=256 dim, K=512 max_seq_len
//   scale = (S*B, D=256) x (D=256, K=512) f32 GEMM  -> V_WMMA_F32_16X16X4_F32
//   scores[b,k,s] = bit(b,k,s) * exp(x - max_s x) / (S1 + 1e-10*S0)
//   (softmax normalizer cancels; only edge/non-edge partial sums needed)
// Everything (64MiB scale + 64MiB out + 2MiB bitmap) is L2-resident (192MB).
// ---------------------------------------------------------------------------

typedef float v2f __attribute__((ext_vector_type(2)));
typedef float v8f __attribute__((ext_vector_type(8)));

#define SEQ   512
#define BATCH 64
#define DIM   256
#define KDIM  512
#define NROWS (SEQ * BATCH)   // 32768 rows of the GEMM (i = s*64 + b)

// ---------------- bitmap helpers ----------------
__global__ void zero_kernel(unsigned* __restrict__ p, int n) {
    int i = blockIdx.x * blockDim.x + threadIdx.x;
    if (i < n) p[i] = 0u;
}

__global__ void edge_kernel(const int* __restrict__ eb, const int* __restrict__ es,
                            const int* __restrict__ ed, unsigned* __restrict__ bitmap, int n) {
    int e = blockIdx.x * blockDim.x + threadIdx.x;
    if (e >= n) return;
    // linear bit index into (b, src=k, dst=s) = (64, 512, 512)
    unsigned linear = (unsigned)eb[e] * (KDIM * SEQ) + (unsigned)es[e] * SEQ + (unsigned)ed[e];
    atomicOr(&bitmap[linear >> 5], 1u << (linear & 31u));
}

// ---------------- f32 WMMA GEMM: scale[i,k] = sum_d M[i,d] * W[k,d] ----------------
// Per wave: 32x64 tile of C = 2 (m) x 4 (n) accumulators, K-step of 4.
// A frag (16x4 f32): lane<16 holds (m=lane, k=0,1), lane>=16 holds (m=lane-16, k=2,3)
//   -> float2 load at M[(i0+lm)*256 + d0 + half*2]
// B frag (4x16 f32): VGPR0 = row k=0 (lanes0-15)/k=2 (lanes16-31), VGPR1 = k=1/k=3
//   -> W is (K,D) row-major == B^T, so float2 load at W[(n0+lm)*256 + d0 + half*2]
__global__ __launch_bounds__(256) void gemm_scale_kernel(const float* __restrict__ M,
                                                         const float* __restrict__ W,
                                                         float* __restrict__ scale) {
    const int lane = threadIdx.x & 31;
    const int wave = threadIdx.x >> 5;          // 8 waves: 2 (row) x 4 (col)
    const int half = lane >> 4;                 // 0: k pair {0,1}, 1: {2,3}
    const int lm   = lane & 15;

    const int i0 = blockIdx.y * 64  + (wave >> 2) * 32;   // row block (32 rows/wave)
    const int n0 = blockIdx.x * 256 + (wave & 3) * 64;    // col block (64 cols/wave)

    v8f acc[2][4] = {};

    const float* Abase = M + (size_t)(i0 + lm) * DIM + half * 2;
    const float* Bbase = W + (size_t)(n0 + lm) * DIM + half * 2;

    for (int d0 = 0; d0 < DIM; d0 += 4) {
        v2f afrag[2], bfrag[4];
#pragma unroll
        for (int mi = 0; mi < 2; ++mi)
            afrag[mi] = *(const v2f*)(Abase + (size_t)mi * 16 * DIM + d0);
#pragma unroll
        for (int ni = 0; ni < 4; ++ni)
            bfrag[ni] = *(const v2f*)(Bbase + (size_t)ni * 16 * DIM + d0);
#pragma unroll
        for (int mi = 0; mi < 2; ++mi)
#pragma unroll
            for (int ni = 0; ni < 4; ++ni)
                acc[mi][ni] = __builtin_amdgcn_wmma_f32_16x16x4_f32(
                    /*neg_a=*/false, afrag[mi],
                    /*neg_b=*/false, bfrag[ni],
                    /*c_mod=*/(short)0, acc[mi][ni],
                    /*reuse_a=*/false, /*reuse_b=*/false);
    }

    // C layout: lanes0-15 -> N=lm, rows v=0..7; lanes16-31 -> N=lm, rows 8+v
#pragma unroll
    for (int mi = 0; mi < 2; ++mi)
#pragma unroll
        for (int ni = 0; ni < 4; ++ni)
#pragma unroll
            for (int v = 0; v < 8; ++v) {
                int m = mi * 16 + half * 8 + v;
                int n = ni * 16 + lm;
                scale[(size_t)(i0 + m) * KDIM + (n0 + n)] = acc[mi][ni][v];
            }
}

// ---------------- fused softmax + edge mask + renorm ----------------
// One thread owns one (b, k) column; 128 blocks x 256 threads = 32768 columns.
// scale layout: scale[(s*64+b)*512 + k] -> lanes (consecutive k) give 128B reads.
__global__ __launch_bounds__(256) void softmax_mask_kernel(const float* __restrict__ scale,
                                                           const unsigned* __restrict__ bitmap,
                                                           float* __restrict__ out) {
    const int b = blockIdx.x >> 1;
    const int k = ((blockIdx.x & 1) << 8) + threadIdx.x;

    const float* col = scale + (size_t)b * KDIM + k;     // element s at col[s*BATCH*KDIM]
    const size_t stride = (size_t)BATCH * KDIM;          // 32768 floats

    // pass 1: max over s (numerical stability, matches jax softmax)
    float mx = -INFINITY;
    for (int s = 0; s < SEQ; ++s)
        mx = fmaxf(mx, col[(size_t)s * stride]);

    // pass 2: edge / non-edge partial sums of exp(x - mx)
    const unsigned* bm = bitmap + ((size_t)b * KDIM + k) * (SEQ / 32);
    float s1 = 0.f, s0 = 0.f;
    for (int w32 = 0; w32 < SEQ / 32; ++w32) {
        unsigned w = bm[w32];
        for (int j = 0; j < 32; ++j) {
            float e = __expf(col[(size_t)(w32 * 32 + j) * stride] - mx);
            if ((w >> j) & 1u) s1 += e; else s0 += e;
        }
    }
    const float inv = 1.0f / (s1 + 1e-10f * s0);

    // pass 3: write scores[b,k,s]
    float* orow = out + ((size_t)b * KDIM + k) * SEQ;
    for (int w32 = 0; w32 < SEQ / 32; ++w32) {
        unsigned w = bm[w32];
        for (int j = 0; j < 32; ++j) {
            int s = w32 * 32 + j;
            float e = __expf(col[(size_t)s * stride] - mx);
            orow[s] = ((w >> j) & 1u) ? e * inv : 0.f;
        }
    }
}

// ---------------------------------------------------------------------------
extern "C" void kernel_launch(void* const* d_in, const int* in_sizes, int n_in,
                              void* d_out, int out_size, void* d_ws, size_t ws_size,
                              hipStream_t stream) {
    const float* M  = (const float*)d_in[0];   // (512, 64, 256) f32
    const float* W  = (const float*)d_in[1];   // (512, 256)     f32
    const int*   eb = (const int*)d_in[2];     // (N_EDGES,)
    const int*   es = (const int*)d_in[3];
    const int*   ed = (const int*)d_in[4];
    // d_in[5] = lengths: unused by the reference
    float* out = (float*)d_out;                // (64, 512, 512) f32

    float*    scale  = (float*)d_ws;                                        // 64 MiB
    unsigned* bitmap = (unsigned*)((char*)d_ws + (size_t)NROWS * KDIM * 4); // +2 MiB

    const int nbm_words = BATCH * KDIM * SEQ / 32;  // 524288
    zero_kernel<<<(nbm_words + 255) / 256, 256, 0, stream>>>(bitmap, nbm_words);

    const int ne = in_sizes[2];
    edge_kernel<<<(ne + 255) / 256, 256, 0, stream>>>(eb, es, ed, bitmap, ne);

    dim3 ggrid(KDIM / 256, NROWS / 64);  // (2, 512) blocks of 8 waves
    gemm_scale_kernel<<<ggrid, 256, 0, stream>>>(M, W, scale);

    softmax_mask_kernel<<<BATCH * 2, 256, 0, stream>>>(scale, bitmap, out);
}